// RoIAlign3D_89481348645402
// MI455X (gfx1250) — compile-verified
//
#include <hip/hip_runtime.h>

typedef float v2f __attribute__((ext_vector_type(2)));
typedef float v8f __attribute__((ext_vector_type(8)));

#define NIMG 4
#define NCH  64
#define DW   64
#define DL   64
#define DH   64
#define CROP 8

// One workgroup = (box, group of 8 (x,y) output columns).
// Per iteration: build R[2][64ch][64h] in LDS (bilinear-xy combined H-rows),
// then z-resample via v_wmma_f32_16x16x4_f32: out[16c,8z] = R[16c,64h] x WzT[64h,8z].
__global__ __launch_bounds__(256) void roialign3d_wmma_kernel(
    const float* __restrict__ feat,   // (4,64,64,64,64) N,C,W,L,H
    const float* __restrict__ boxes,  // (512,6)
    const int*   __restrict__ box_ind,// (512,)
    float*       __restrict__ out)    // (512,64,8,8,8)
{
    __shared__ float R[2][64][68];    // 2 columns x 64 ch x 64 h (+4 pad): 34816 B

    const int box    = blockIdx.x >> 3;
    const int colgrp = blockIdx.x & 7;
    const int tid    = threadIdx.x;
    const int lane   = tid & 31;
    const int wave   = tid >> 5;

    const float bx1 = boxes[box*6+0], by1 = boxes[box*6+1], bz1 = boxes[box*6+2];
    const float bx2 = boxes[box*6+3], by2 = boxes[box*6+4], bz2 = boxes[box*6+5];
    const int   n   = box_ind[box];

    const float sx = (bx2-bx1)*63.0f/7.0f;
    const float sy = (by2-by1)*63.0f/7.0f;
    const float sz = (bz2-bz1)*63.0f/7.0f;

    // ---------------- B operand: Wz in WMMA 4x16 striping, box-constant -------------
    // B[k,n] layout: VGPR0 = rows k=0 (lanes0-15) / k=2 (lanes16-31); VGPR1 = k=1 / k=3.
    // n = z sample (only 0..7 used), k enumerates h within the K=4 step.
    const int m16 = lane & 15;   // z for B/D, channel-row for A
    const int hlf = lane >> 4;   // 0: k,k+1 half ; 1: k+2,k+3 half
    {
    }
    const float fz   = bz1*63.0f + sz * (float)m16;
    const bool  zok  = (m16 < CROP) && (fz >= 0.0f) && (fz <= 63.0f);
    const float fzl  = floorf(fz);
    const float zfr  = fz - fzl;
    int zlo = (int)fzl; zlo = zlo < 0 ? 0 : (zlo > 63 ? 63 : zlo);
    const int zhi = (zlo + 1 > 63) ? 63 : zlo + 1;
    const float wz0 = zok ? (1.0f - zfr) : 0.0f;
    const float wz1 = zok ? zfr          : 0.0f;

    v2f Bv[16];
#pragma unroll
    for (int s = 0; s < 16; ++s) {
        const int h0 = 4*s + 2*hlf;
        const int h1 = h0 + 1;
        Bv[s].x = (h0 == zlo ? wz0 : 0.0f) + (h0 == zhi ? wz1 : 0.0f);
        Bv[s].y = (h1 == zlo ? wz0 : 0.0f) + (h1 == zhi ? wz1 : 0.0f);
    }

    // R-build mapping: 128 threads per column; 16 threads sweep h (float4), 8 ch-phases.
    const int rcol = tid >> 7;      // which of the 2 columns
    const int rloc = tid & 127;
    const int rh16 = rloc & 15;     // h/4
    const int rc8  = rloc >> 4;     // channel phase base 0..7

    // WMMA mapping: wave -> (column, 16-channel tile)
    const int wcol = wave >> 2;         // 0..1
    const int c0   = (wave & 3) * 16;   // channel tile base

    for (int it = 0; it < 4; ++it) {
        const int q0 = colgrp*8 + it*2;

        // -------- build R for two (x,y) columns: 4 corner H-rows, bilinear combine ----
        {
            const int q = q0 + rcol;
            const int x = q >> 3, y = q & 7;
            const float fx = bx1*63.0f + sx * (float)x;
            const float fy = by1*63.0f + sy * (float)y;
            const bool xok = (fx >= 0.0f) && (fx <= 63.0f);
            const bool yok = (fy >= 0.0f) && (fy <= 63.0f);
            const float fxl = floorf(fx), fyl = floorf(fy);
            const float xf = fx - fxl,   yf = fy - fyl;
            int xlo = (int)fxl; xlo = xlo < 0 ? 0 : (xlo > 63 ? 63 : xlo);
            int ylo = (int)fyl; ylo = ylo < 0 ? 0 : (ylo > 63 ? 63 : ylo);
            const int xhi = (xlo + 1 > 63) ? 63 : xlo + 1;
            const int yhi = (ylo + 1 > 63) ? 63 : ylo + 1;
            const float wx0 = xok ? (1.0f - xf) : 0.0f, wx1 = xok ? xf : 0.0f;
            const float wy0 = yok ? (1.0f - yf) : 0.0f, wy1 = yok ? yf : 0.0f;
            const float w00 = wx0*wy0, w01 = wx0*wy1, w10 = wx1*wy0, w11 = wx1*wy1;

#pragma unroll
            for (int i = 0; i < 8; ++i) {
                const int c = rc8 + 8*i;
                const long b = (long)(n*NCH + c) * DW;
                const float4* p00 = (const float4*)(feat + ((b + xlo)*DL + ylo)*DH) + rh16;
                const float4* p01 = (const float4*)(feat + ((b + xlo)*DL + yhi)*DH) + rh16;
                const float4* p10 = (const float4*)(feat + ((b + xhi)*DL + ylo)*DH) + rh16;
                const float4* p11 = (const float4*)(feat + ((b + xhi)*DL + yhi)*DH) + rh16;
                const float4 a = *p00, bb = *p01, cc = *p10, dd = *p11;
                float4 r;
                r.x = w00*a.x + w01*bb.x + w10*cc.x + w11*dd.x;
                r.y = w00*a.y + w01*bb.y + w10*cc.y + w11*dd.y;
                r.z = w00*a.z + w01*bb.z + w10*cc.z + w11*dd.z;
                r.w = w00*a.w + w01*bb.w + w10*cc.w + w11*dd.w;
                *(float4*)&R[rcol][c][4*rh16] = r;
            }
        }
        __syncthreads();

        // -------- z-contraction: 16 x v_wmma_f32_16x16x4_f32 per wave ---------------
        {
            v8f acc = {0.0f,0.0f,0.0f,0.0f,0.0f,0.0f,0.0f,0.0f};
            // A 16x4 striping: lanes0-15 & 16-31 both hold rows m=0..15;
            // VGPR0 = k / k+2, VGPR1 = k+1 / k+3  -> v2f at h = 4s + 2*hlf.
            const float* rrow = &R[wcol][c0 + m16][0];
#pragma unroll
            for (int s = 0; s < 16; ++s) {
                const v2f a = *(const v2f*)(rrow + 4*s + 2*hlf);
                acc = __builtin_amdgcn_wmma_f32_16x16x4_f32(
                        /*neg_a=*/false, a, /*neg_b=*/false, Bv[s],
                        /*c_mod=*/(short)0, acc, /*reuse_a=*/false, /*reuse_b=*/false);
            }
            // D layout: lane -> n=z (= m16), VGPR r -> channel row r (lanes<16) / r+8.
            const int q = q0 + wcol;
            const int x = q >> 3, y = q & 7;
            const int z = m16;
            if (z < CROP) {
#pragma unroll
                for (int r = 0; r < 8; ++r) {
                    const int c = c0 + r + hlf*8;
                    out[((long)(box*NCH + c) << 9) + x*64 + y*8 + z] = acc[r];
                }
            }
        }
        __syncthreads();
    }
}

extern "C" void kernel_launch(void* const* d_in, const int* in_sizes, int n_in,
                              void* d_out, int out_size, void* d_ws, size_t ws_size,
                              hipStream_t stream) {
    const float* feat    = (const float*)d_in[0];
    const float* boxes   = (const float*)d_in[1];
    const int*   box_ind = (const int*)d_in[2];
    float*       out     = (float*)d_out;
    (void)in_sizes; (void)n_in; (void)out_size; (void)d_ws; (void)ws_size;

    dim3 grid(512 * 8);   // 512 boxes x 8 column-groups
    dim3 block(256);      // 8 wave32
    roialign3d_wmma_kernel<<<grid, block, 0, stream>>>(feat, boxes, box_ind, out);
}